// _NeuronWilsonDslashAdapter_48739288875229
// MI455X (gfx1250) — compile-verified
//
#include <hip/hip_runtime.h>

typedef float v2f __attribute__((ext_vector_type(2)));
typedef float v4f __attribute__((ext_vector_type(4)));
typedef float v8f __attribute__((ext_vector_type(8)));

#define VOL (32 * 32 * 32 * 32)

// Wave-cooperative Wilson Dslash:
//   wave = 32 lanes -> 16 sites; lane = (site n = lane%16, spin-half kh = lane/16)
//   VALU: SU(3) color matvecs (per-site U, not shareable)
//   WMMA: shared 4x4 spin projectors as A (Pr rows 0-3, Pi rows 4-7),
//         per-lane matvec results as B (K = spin, N = site), C chained over all 8 hop terms.
__global__ __launch_bounds__(256) void wilson_dslash_wmma(
    const float* __restrict__ psi_re, const float* __restrict__ psi_im,
    const float* __restrict__ U_re,   const float* __restrict__ U_im,
    const float* __restrict__ Pm_re,  const float* __restrict__ Pm_im,
    const float* __restrict__ Pp_re,  const float* __restrict__ Pp_im,
    float* __restrict__ out_re,       float* __restrict__ out_im)
{
    const int tid  = blockIdx.x * blockDim.x + threadIdx.x;
    const int lane = tid & 31;
    const int wv   = tid >> 5;
    const int n    = lane & 15;   // site within wave's 16-site block; also A-matrix row M
    const int kh   = lane >> 4;   // spin half: 0 -> spins {0,1}, 1 -> spins {2,3}
    const int site = wv * 16 + n; // exact grid: site < VOL always

    const int x3 = site & 31, x2 = (site >> 5) & 31, x1 = (site >> 10) & 31, x0 = (site >> 15) & 31;
    const int coords[4]  = { x0, x1, x2, x3 };
    const int strides[4] = { 32768, 1024, 32, 1 };

    __builtin_prefetch(U_re  + (long)site * 36, 0, 3);
    __builtin_prefetch(psi_re + (long)site * 12, 0, 3);

    // ---- Build WMMA A-operands for all 4 directions -------------------------
    // A is 16x4 (M x K): lane gives M = n, lane group gives K-half.
    // Rows 0-3 = P_re, rows 4-7 = P_im, rows 8-15 = 0  (one WMMA => Pr*B and Pi*B).
    float AmK0[4], AmK1[4], ApK0[4], ApK1[4];
    {
        const int k0 = 2 * kh, k1 = 2 * kh + 1;
#pragma unroll
        for (int mu = 0; mu < 4; ++mu) {
            float m0 = 0.f, m1 = 0.f, p0 = 0.f, p1 = 0.f;
            if (n < 4) {
                const int b = mu * 16 + n * 4;
                m0 = Pm_re[b + k0]; m1 = Pm_re[b + k1];
                p0 = Pp_re[b + k0]; p1 = Pp_re[b + k1];
            } else if (n < 8) {
                const int b = mu * 16 + (n - 4) * 4;
                m0 = Pm_im[b + k0]; m1 = Pm_im[b + k1];
                p0 = Pp_im[b + k0]; p1 = Pp_im[b + k1];
            }
            AmK0[mu] = m0; AmK1[mu] = m1;
            ApK0[mu] = p0; ApK1[mu] = p1;
        }
    }

    // Accumulators: per color i, acc1 = sum_t (A_t x W_re_t), acc2 = sum_t (A_t x W_im_t)
    v8f acc1[3], acc2[3];
#pragma unroll
    for (int i = 0; i < 3; ++i) { acc1[i] = (v8f){}; acc2[i] = (v8f){}; }

#pragma unroll
    for (int mu = 0; mu < 4; ++mu) {
        const int st = strides[mu], c = coords[mu];
        const int nf = (c == 31) ? site - 31 * st : site + st; // x + mu (periodic)
        const int nb = (c == 0)  ? site + 31 * st : site - st; // x - mu (periodic)
        const v2f Am = { AmK0[mu], AmK1[mu] };
        const v2f Ap = { ApK0[mu], ApK1[mu] };

        // ================= forward hop: Pm_mu * ( U_mu(x) * psi(x+mu) ) =====
        {
            float ur[9], ui[9];
            const long ub = (long)site * 36 + mu * 9;
#pragma unroll
            for (int q = 0; q < 9; ++q) { ur[q] = U_re[ub + q]; ui[q] = U_im[ub + q]; }

            float pr[6], pi[6]; // spins {2kh, 2kh+1} x 3 colors, 8B-aligned float2 loads
            const long pb = (long)nf * 12 + kh * 6;
#pragma unroll
            for (int q = 0; q < 3; ++q) {
                v2f a = *(const v2f*)(psi_re + pb + 2 * q);
                v2f b = *(const v2f*)(psi_im + pb + 2 * q);
                pr[2 * q] = a.x; pr[2 * q + 1] = a.y;
                pi[2 * q] = b.x; pi[2 * q + 1] = b.y;
            }

            float wr[2][3], wi[2][3]; // W = U * psi_halfspinor (complex)
#pragma unroll
            for (int s = 0; s < 2; ++s)
#pragma unroll
                for (int i = 0; i < 3; ++i) {
                    float ar = 0.f, ai = 0.f;
#pragma unroll
                    for (int j = 0; j < 3; ++j) {
                        const float urx = ur[i * 3 + j], uix = ui[i * 3 + j];
                        const float vr = pr[s * 3 + j],  vi = pi[s * 3 + j];
                        ar = fmaf(urx, vr, ar); ar = fmaf(-uix, vi, ar);
                        ai = fmaf(urx, vi, ai); ai = fmaf( uix, vr, ai);
                    }
                    wr[s][i] = ar; wi[s][i] = ai;
                }
#pragma unroll
            for (int i = 0; i < 3; ++i) {
                const v2f br = { wr[0][i], wr[1][i] };
                const v2f bi = { wi[0][i], wi[1][i] };
                acc1[i] = __builtin_amdgcn_wmma_f32_16x16x4_f32(
                    false, Am, false, br, (short)0, acc1[i], false, false);
                acc2[i] = __builtin_amdgcn_wmma_f32_16x16x4_f32(
                    false, Am, false, bi, (short)0, acc2[i], false, false);
            }
        }

        // ============ backward hop: Pp_mu * ( U_mu(x-mu)^dag * psi(x-mu) ) ==
        {
            float ur[9], ui[9];
            const long ub = (long)nb * 36 + mu * 9;
#pragma unroll
            for (int q = 0; q < 9; ++q) { ur[q] = U_re[ub + q]; ui[q] = U_im[ub + q]; }

            float pr[6], pi[6];
            const long pb = (long)nb * 12 + kh * 6;
#pragma unroll
            for (int q = 0; q < 3; ++q) {
                v2f a = *(const v2f*)(psi_re + pb + 2 * q);
                v2f b = *(const v2f*)(psi_im + pb + 2 * q);
                pr[2 * q] = a.x; pr[2 * q + 1] = a.y;
                pi[2 * q] = b.x; pi[2 * q + 1] = b.y;
            }

            float wr[2][3], wi[2][3]; // W = U^dagger * psi_halfspinor
#pragma unroll
            for (int s = 0; s < 2; ++s)
#pragma unroll
                for (int i = 0; i < 3; ++i) {
                    float ar = 0.f, ai = 0.f;
#pragma unroll
                    for (int j = 0; j < 3; ++j) {
                        const float urx = ur[j * 3 + i], uix = ui[j * 3 + i]; // transpose
                        const float vr = pr[s * 3 + j],  vi = pi[s * 3 + j];
                        ar = fmaf(urx, vr, ar); ar = fmaf( uix, vi, ar);     // conj
                        ai = fmaf(urx, vi, ai); ai = fmaf(-uix, vr, ai);
                    }
                    wr[s][i] = ar; wi[s][i] = ai;
                }
#pragma unroll
            for (int i = 0; i < 3; ++i) {
                const v2f br = { wr[0][i], wr[1][i] };
                const v2f bi = { wi[0][i], wi[1][i] };
                acc1[i] = __builtin_amdgcn_wmma_f32_16x16x4_f32(
                    false, Ap, false, br, (short)0, acc1[i], false, false);
                acc2[i] = __builtin_amdgcn_wmma_f32_16x16x4_f32(
                    false, Ap, false, bi, (short)0, acc2[i], false, false);
            }
        }
    }

    // ---- Epilogue: complex recombine + (-0.5), lanes 0-15 own rows 0-7 -----
    // D vgpr v, lanes 0-15 => row M=v, N=n.  rows 0-3 = Pr*W, rows 4-7 = Pi*W.
    //   out_re = -0.5*(acc1.top - acc2.bot),  out_im = -0.5*(acc2.top + acc1.bot)
    if (kh == 0) {
        float r[12], m[12];
#pragma unroll
        for (int s = 0; s < 4; ++s)
#pragma unroll
            for (int i = 0; i < 3; ++i) {
                r[s * 3 + i] = -0.5f * (acc1[i][s] - acc2[i][s + 4]);
                m[s * 3 + i] = -0.5f * (acc2[i][s] + acc1[i][s + 4]);
            }
        const long ob = (long)site * 12; // 48B => 16B aligned: float4 stores
#pragma unroll
        for (int q = 0; q < 3; ++q) {
            *(v4f*)(out_re + ob + 4 * q) = (v4f){ r[4*q], r[4*q+1], r[4*q+2], r[4*q+3] };
            *(v4f*)(out_im + ob + 4 * q) = (v4f){ m[4*q], m[4*q+1], m[4*q+2], m[4*q+3] };
        }
    }
}

extern "C" void kernel_launch(void* const* d_in, const int* in_sizes, int n_in,
                              void* d_out, int out_size, void* d_ws, size_t ws_size,
                              hipStream_t stream) {
    const float* psi_re = (const float*)d_in[0];
    const float* psi_im = (const float*)d_in[1];
    const float* U_re   = (const float*)d_in[2];
    const float* U_im   = (const float*)d_in[3];
    const float* Pm_re  = (const float*)d_in[4];
    const float* Pm_im  = (const float*)d_in[5];
    const float* Pp_re  = (const float*)d_in[6];
    const float* Pp_im  = (const float*)d_in[7];

    float* out_re = (float*)d_out;
    float* out_im = out_re + (size_t)VOL * 12;

    // 16 sites per wave32, 8 waves per 256-thread block -> 128 sites/block
    const int blocks = VOL / 128; // 8192, exact cover (EXEC all-ones for WMMA)
    wilson_dslash_wmma<<<blocks, 256, 0, stream>>>(
        psi_re, psi_im, U_re, U_im, Pm_re, Pm_im, Pp_re, Pp_im, out_re, out_im);
}